// CenterLoss_33844342293038
// MI455X (gfx1250) — compile-verified
//
#include <hip/hip_runtime.h>

typedef __attribute__((ext_vector_type(2))) float v2f;
typedef __attribute__((ext_vector_type(8))) float v8f;

#define FEAT_DIM 512
#define WAVES_PER_BLOCK 4
#define K_PER_WAVE (FEAT_DIM / WAVES_PER_BLOCK)   // 128 -> 32 wmma k-chunks of 4

// One block = one 16-sample tile, 4 waves each covering a 128-wide K slice.
// Each wave computes diag(F F^T), diag(C C^T), diag(F C^T) for its K slice via
// V_WMMA_F32_16X16X4_F32, using the fact that the per-lane register layout of a
// 16x4 f32 A-operand of matrix X is identical to the 4x16 B-operand of X^T:
// lane l holds row (l&15), K = k0 + 2*(l>>4) + {0,1}.
__global__ __launch_bounds__(128) void center_loss_tile_kernel(
    const float* __restrict__ features,
    const int*   __restrict__ labels,
    const float* __restrict__ centers,
    float*       __restrict__ partial)
{
    __shared__ float s_fsq[WAVES_PER_BLOCK][16];
    __shared__ float s_csq[WAVES_PER_BLOCK][16];
    __shared__ float s_dot[WAVES_PER_BLOCK][16];

    const int tile  = blockIdx.x;
    const int tid   = threadIdx.x;
    const int wave  = tid >> 5;
    const int lane  = tid & 31;
    const int row   = lane & 15;          // M (and N) index handled by this lane
    const int khalf = (lane >> 4) << 1;   // 0 for lanes 0-15, 2 for lanes 16-31

    const int sample = tile * 16 + row;
    const int lab    = labels[sample];

    const float* f_base = features + (size_t)sample * FEAT_DIM + (size_t)(wave * K_PER_WAVE + khalf);
    const float* c_base = centers  + (size_t)lab    * FEAT_DIM + (size_t)(wave * K_PER_WAVE + khalf);

    v8f accFF = {};   // diag -> ||f||^2 partial
    v8f accCC = {};   // diag -> ||c||^2 partial
    v8f accFC = {};   // diag -> f . c   partial

    #pragma unroll 8
    for (int k = 0; k < K_PER_WAVE; k += 4) {
        v2f f = *(const v2f*)(f_base + k);
        v2f c = *(const v2f*)(c_base + k);
        // D = A x B + C ; same registers serve as A (of X) and B (of X^T)
        accFC = __builtin_amdgcn_wmma_f32_16x16x4_f32(false, f, false, c, (short)0, accFC, false, false);
        accFF = __builtin_amdgcn_wmma_f32_16x16x4_f32(false, f, false, f, (short)0, accFF, false, false);
        accCC = __builtin_amdgcn_wmma_f32_16x16x4_f32(false, c, false, c, (short)0, accCC, false, false);
    }

    // Diagonal extraction per documented 16x16 f32 C/D layout:
    //   m in 0..7  : lane m,      acc element m
    //   m in 8..15 : lane m+16,   acc element m-8
    const bool has = (lane < 8) || (lane >= 24);
    const int  m   = (lane < 8) ? lane : (lane - 16);
    const int  idx = lane & 7;
    if (has) {
        s_fsq[wave][m] = accFF[idx];
        s_csq[wave][m] = accCC[idx];
        s_dot[wave][m] = accFC[idx];
    }
    __syncthreads();

    if (tid < 32) {
        float val = 0.0f;
        if (tid < 16) {
            float fsq = 0.0f, csq = 0.0f, dt = 0.0f;
            #pragma unroll
            for (int w = 0; w < WAVES_PER_BLOCK; ++w) {
                fsq += s_fsq[w][tid];
                csq += s_csq[w][tid];
                dt  += s_dot[w][tid];
            }
            float dist = fsq + csq - 2.0f * dt;
            val = fminf(fmaxf(dist, 1e-12f), 1e12f);
        }
        #pragma unroll
        for (int off = 16; off > 0; off >>= 1)
            val += __shfl_down(val, off, 32);
        if (tid == 0)
            partial[tile] = val;
    }
}

// Deterministic single-wave final reduction: fixed summation structure, no
// float atomics, overwrites (not accumulates into) the poisoned d_out.
__global__ __launch_bounds__(32) void center_loss_reduce_kernel(
    const float* __restrict__ partial,
    float*       __restrict__ out,
    int ntiles, int batch, int num_classes)
{
    const int lane = threadIdx.x;
    float s = 0.0f;
    for (int i = lane; i < ntiles; i += 32)
        s += partial[i];
    #pragma unroll
    for (int off = 16; off > 0; off >>= 1)
        s += __shfl_down(s, off, 32);
    if (lane == 0) {
        // masked-out zeros clamp to 1e-12: B*(C-1) of them, divided by B
        out[0] = s / (float)batch + (float)(num_classes - 1) * 1e-12f;
    }
}

extern "C" void kernel_launch(void* const* d_in, const int* in_sizes, int n_in,
                              void* d_out, int out_size, void* d_ws, size_t ws_size,
                              hipStream_t stream) {
    const float* features = (const float*)d_in[0];
    const int*   labels   = (const int*)d_in[1];
    const float* centers  = (const float*)d_in[2];
    float*       out      = (float*)d_out;
    float*       partial  = (float*)d_ws;       // ntiles floats of scratch

    const int batch       = in_sizes[1];                 // 4096
    const int num_classes = in_sizes[2] / FEAT_DIM;      // 10000
    const int ntiles      = batch / 16;                  // 256

    hipLaunchKernelGGL(center_loss_tile_kernel, dim3(ntiles), dim3(128), 0, stream,
                       features, labels, centers, partial);
    hipLaunchKernelGGL(center_loss_reduce_kernel, dim3(1), dim3(32), 0, stream,
                       partial, out, ntiles, batch, num_classes);
}